// EarthMoverDistance_56616258895920
// MI455X (gfx1250) — compile-verified
//
#include <hip/hip_runtime.h>
#include <math.h>

typedef float v2f __attribute__((ext_vector_type(2)));
typedef float v8f __attribute__((ext_vector_type(8)));

#define SK_EPS     0.05f
#define SK_NITER   30
#define SK_N       2048
#define SK_NB      8
#define SK_NPROB   24                  // 8 batches * {xy, xx, yy}
// 20 * log2(e): converts (u - c)/eps into base-2 exponent domain
#define SK_C2      28.853900817779268f
// eps * ln(2): converts base-2 LSE back to natural log domain
#define SK_EPSLN2  0.034657359027997264f
// eps * log(2048)
#define SK_ADDC    0.3812309471549990f

// Workspace layout (floats):
//   f   : [24][2048]                    offset 0
//   g   : [24][2048]                    offset 24*2048
//   pk1 : [8][2048] float4 (x,y,z,n)    offset 48*2048
//   pk2 : [8][2048] float4 (x,y,z,n)    offset 48*2048 + 8*2048*4
// Total = 229376 floats = 896 KB.

__global__ void sk_prep_kernel(const float* __restrict__ xyz1,
                               const float* __restrict__ xyz2,
                               float4* __restrict__ pk1, float4* __restrict__ pk2) {
  int idx = blockIdx.x * blockDim.x + threadIdx.x;
  if (idx < SK_NB * SK_N) {
    float a0 = xyz1[3 * idx + 0], a1 = xyz1[3 * idx + 1], a2 = xyz1[3 * idx + 2];
    pk1[idx] = make_float4(a0, a1, a2, a0 * a0 + a1 * a1 + a2 * a2);
    float b0 = xyz2[3 * idx + 0], b1 = xyz2[3 * idx + 1], b2 = xyz2[3 * idx + 2];
    pk2[idx] = make_float4(b0, b1, b2, b0 * b0 + b1 * b1 + b2 * b2);
  }
}

__global__ void sk_zero_kernel(float* __restrict__ p, int n) {
  int idx = blockIdx.x * blockDim.x + threadIdx.x;
  if (idx < n) p[idx] = 0.0f;
}

// One log-sum-exp half-iteration. Rows = point set A, reduce over point set B
// with dual potential u over B; writes v over A:
//   v_i = EPS*log(N) - EPS * LSE_j( (u_j - dist(A_i,B_j)) / EPS )
// grid: x = row tile (SK_N/64 = 32), y = problem (24). block = 128 (4 wave32).
__global__ __launch_bounds__(128)
void sk_lse_pass_kernel(float* __restrict__ ws, int updateF) {
  const int prob = blockIdx.y;
  const int b = prob / 3;
  const int t = prob - 3 * b;     // 0=xy, 1=xx, 2=yy

  float* f = ws;
  float* g = ws + SK_NPROB * SK_N;
  const float4* pk1 = (const float4*)(ws + 2 * SK_NPROB * SK_N) + (size_t)b * SK_N;
  const float4* pk2 = (const float4*)(ws + 2 * SK_NPROB * SK_N) + (size_t)(SK_NB + b) * SK_N;

  const float4 *pkA, *pkB;
  if (t == 0) {
    if (updateF) { pkA = pk1; pkB = pk2; }
    else         { pkA = pk2; pkB = pk1; }
  } else if (t == 1) { pkA = pk1; pkB = pk1; }
  else               { pkA = pk2; pkB = pk2; }

  const float* u    = updateF ? (g + prob * SK_N) : (f + prob * SK_N);
  float*       vout = updateF ? (f + prob * SK_N) : (g + prob * SK_N);

  // Stage B-side data in LDS: (x,y,z,|y|^2) + pre-scaled potential. 40 KB.
  __shared__ float4 sB[SK_N];
  __shared__ float  sU[SK_N];
  for (int j = threadIdx.x; j < SK_N; j += 128) {
    sB[j] = pkB[j];
    sU[j] = u[j] * SK_C2;     // pre-scale into base-2 exponent units
  }
  __syncthreads();

  const int lane = threadIdx.x & 31;
  const int wave = threadIdx.x >> 5;    // 0..3
  const int half = lane >> 4;           // 0 or 1
  const int lm   = lane & 15;
  const int rowBase = blockIdx.x * 64 + wave * 16;

  // A operand: 16x4 f32, lanes 0-15 hold M=lane {K0,K1}, lanes 16-31 hold {K2,K3}.
  const float4 pa = pkA[rowBase + lm];
  v2f aop;
  aop.x = half ? pa.z : pa.x;
  aop.y = half ? 0.0f : pa.y;

  // Row norms for the 8 D-rows this lane owns (M = v + 8*half).
  float nA8[8];
#pragma unroll
  for (int v = 0; v < 8; ++v) nA8[v] = pkA[rowBase + half * 8 + v].w;

  float mrun[8], srun[8];
#pragma unroll
  for (int v = 0; v < 8; ++v) { mrun[v] = -__builtin_huge_valf(); srun[v] = 0.0f; }

  for (int ct = 0; ct < SK_N / 16; ++ct) {
    const int col = ct * 16 + lm;
    const float4 bp = sB[col];
    const float  uc2 = sU[col];
    v2f bop;                       // B: 4x16 f32, lanes hold N=lm, VGPR/half pick K
    bop.x = half ? bp.z : bp.x;
    bop.y = half ? 0.0f : bp.y;
    const float nBc = bp.w;

    v8f acc = {};
    v8f d = __builtin_amdgcn_wmma_f32_16x16x4_f32(
        /*neg_a=*/false, aop, /*neg_b=*/false, bop,
        /*c_mod=*/(short)0, acc, /*reuse_a=*/false, /*reuse_b=*/false);

#pragma unroll
    for (int v = 0; v < 8; ++v) {
      float d2 = nA8[v] + nBc - 2.0f * d[v];
      float c  = __builtin_amdgcn_sqrtf(fmaxf(d2, 0.0f) + 1e-12f);
      float tt = fmaf(c, -SK_C2, uc2);            // (u - c)/eps in log2 units
      float mo = mrun[v];
      float mn = fmaxf(mo, tt);
      srun[v] = srun[v] * __builtin_amdgcn_exp2f(mo - mn)
              + __builtin_amdgcn_exp2f(tt - mn);
      mrun[v] = mn;
    }
  }

  // Merge LSE state across the 16 lanes of each half-wave (they share rows).
#pragma unroll
  for (int v = 0; v < 8; ++v) {
    float m = mrun[v], s = srun[v];
#pragma unroll
    for (int off = 1; off < 16; off <<= 1) {
      float mo = __shfl_xor(m, off, 32);
      float so = __shfl_xor(s, off, 32);
      float mn = fmaxf(m, mo);
      s = s * __builtin_amdgcn_exp2f(m - mn) + so * __builtin_amdgcn_exp2f(mo - mn);
      m = mn;
    }
    mrun[v] = m; srun[v] = s;
  }

  // Lane with lm==v (lm<8) in each half writes row rowBase + 8*half + v.
  if (lm < 8) {
    float mv = mrun[0], sv = srun[0];
#pragma unroll
    for (int v = 1; v < 8; ++v) {
      if (lm == v) { mv = mrun[v]; sv = srun[v]; }
    }
    const float lse2 = mv + __builtin_amdgcn_logf(sv);    // v_log_f32 = log2
    vout[rowBase + 8 * half + lm] = SK_ADDC - SK_EPSLN2 * lse2;
  }
}

// out = sum_p w_t * (mean(f_p) + mean(g_p)) / B,  w = (1, -0.5, -0.5)
__global__ void sk_reduce_kernel(const float* __restrict__ ws, float* __restrict__ out) {
  __shared__ float red[256];
  const float* f = ws;
  const float* g = ws + SK_NPROB * SK_N;
  float acc = 0.0f;
  for (int idx = threadIdx.x; idx < SK_NPROB * SK_N; idx += 256) {
    int prob = idx / SK_N;
    int t = prob % 3;
    float w = (t == 0) ? 1.0f : -0.5f;
    acc += w * (f[idx] + g[idx]);
  }
  acc *= (1.0f / SK_N) * (1.0f / SK_NB);
  red[threadIdx.x] = acc;
  __syncthreads();
  for (int s = 128; s > 0; s >>= 1) {
    if (threadIdx.x < s) red[threadIdx.x] += red[threadIdx.x + s];
    __syncthreads();
  }
  if (threadIdx.x == 0) out[0] = red[0];
}

extern "C" void kernel_launch(void* const* d_in, const int* in_sizes, int n_in,
                              void* d_out, int out_size, void* d_ws, size_t ws_size,
                              hipStream_t stream) {
  const float* xyz1 = (const float*)d_in[0];   // (8, 2048, 3) f32
  const float* xyz2 = (const float*)d_in[1];   // (8, 2048, 3) f32
  float* out = (float*)d_out;                  // scalar f32
  float* ws  = (float*)d_ws;

  float4* pk1 = (float4*)(ws + 2 * SK_NPROB * SK_N);
  float4* pk2 = pk1 + (size_t)SK_NB * SK_N;

  sk_prep_kernel<<<(SK_NB * SK_N + 255) / 256, 256, 0, stream>>>(xyz1, xyz2, pk1, pk2);
  sk_zero_kernel<<<(2 * SK_NPROB * SK_N + 255) / 256, 256, 0, stream>>>(ws, 2 * SK_NPROB * SK_N);

  dim3 grid(SK_N / 64, SK_NPROB);
  for (int it = 0; it < SK_NITER; ++it) {
    sk_lse_pass_kernel<<<grid, 128, 0, stream>>>(ws, 1);  // f-update
    sk_lse_pass_kernel<<<grid, 128, 0, stream>>>(ws, 0);  // g-update
  }
  sk_reduce_kernel<<<1, 256, 0, stream>>>(ws, out);
}